// ActorGCN_89928025244585
// MI455X (gfx1250) — compile-verified
//
#include <hip/hip_runtime.h>
#include <hip/hip_bf16.h>
#include <math.h>

// ---------------------------------------------------------------------------
// ActorGCN fused pipeline for gfx1250 (MI455X), compile-only loop.
//
// Key algebraic collapses:
//   * gcn_b cancels in batchnorm -> never added.
//   * BN stats from s20 = sum_m agg_m  and  G = agg^T agg (20x20 Gram),
//     computed with V_WMMA_F32_16X16X4_F32 (fp32 matrix pipe, K=4).
//   * post-BN linear folds into a 20x2 matrix: logits = relu(agg@Wf + const).
// Net effect: the 1.6GB h matrix is never materialized; the kernel is
// aggregation-bound (edges + L2-resident atomics).
//
// Round-2 change: k_gram inner loop is now branch-free — agg is padded with
// 4 zero rows (node-bound guards gone) and the feature-pad guard became a
// clamped-address load + v_cndmask select. Hot loop = 4 coalesced
// global_load_b32 + 4 v_wmma, unrolled x2.
// ---------------------------------------------------------------------------

typedef __attribute__((ext_vector_type(2))) float v2f;
typedef __attribute__((ext_vector_type(8))) float v8f;

#define IN_DIM   20
#define HIDDEN   4096
#define OUT_DIM  2
#define BN_EPS   1e-5f
#define AGG_PAD  4                 // zero rows appended to agg for k_gram

// ---------------- init: deg=1 (self loop), zero G/s20/agg-padding ----------
__global__ void k_init(float* __restrict__ deg, float* __restrict__ G,
                       float* __restrict__ s20, float* __restrict__ aggpad,
                       int n) {
  int i = blockIdx.x * blockDim.x + threadIdx.x;
  if (i < n)                deg[i]    = 1.0f;
  if (i < 32*32)            G[i]      = 0.0f;
  if (i < 32)               s20[i]    = 0.0f;
  if (i < AGG_PAD * IN_DIM) aggpad[i] = 0.0f;
}

// ---------------- degree accumulation over edge cols ------------------------
__global__ void k_deg(const int* __restrict__ ei, float* __restrict__ deg,
                      int nedges) {
  int e = blockIdx.x * blockDim.x + threadIdx.x;
  if (e >= nedges) return;
  int c = ei[nedges + e];          // edge_index[1][e]
  atomicAdd(&deg[c], 1.0f);
}

// ---------------- dinv = rsqrt(deg) ----------------------------------------
__global__ void k_dinv(const float* __restrict__ deg, float* __restrict__ dinv,
                       int n) {
  int i = blockIdx.x * blockDim.x + threadIdx.x;
  if (i >= n) return;
  float d = deg[i];
  dinv[i] = (d > 0.0f) ? rsqrtf(d) : 0.0f;
}

// ---------------- self-loop contribution: agg = nf * dinv^2 ----------------
__global__ void k_selfagg(const float* __restrict__ nf,
                          const float* __restrict__ dinv,
                          float* __restrict__ agg, int total) {
  int t = blockIdx.x * blockDim.x + threadIdx.x;
  if (t >= total) return;
  int i = t / IN_DIM;
  float di = dinv[i];
  agg[t] = nf[t] * (di * di);
}

// ---------------- edge scatter: agg[col] += nf[row] * dinv[r]*dinv[c] ------
__global__ void k_edge(const int* __restrict__ ei,
                       const float* __restrict__ nf,
                       const float* __restrict__ dinv,
                       float* __restrict__ agg, int nedges) {
  int e = blockIdx.x * blockDim.x + threadIdx.x;
  if (e >= nedges) return;
  int r = ei[e];
  int c = ei[nedges + e];
  float nrm = dinv[r] * dinv[c];
  // row stride 20 floats = 80B -> 16B aligned: float4 gathers
  const float4* src = (const float4*)(nf + (size_t)r * IN_DIM);
  float* dst = agg + (size_t)c * IN_DIM;
  #pragma unroll
  for (int q = 0; q < IN_DIM / 4; ++q) {
    float4 v = src[q];
    atomicAdd(dst + 4*q + 0, v.x * nrm);
    atomicAdd(dst + 4*q + 1, v.y * nrm);
    atomicAdd(dst + 4*q + 2, v.z * nrm);
    atomicAdd(dst + 4*q + 3, v.w * nrm);
  }
}

// ---------------- column sums s20[k] = sum_m agg[m][k] (LDS reduce) --------
__global__ void k_s20(const float* __restrict__ agg, float* __restrict__ s20,
                      int n) {
  __shared__ float ls[IN_DIM];
  int tid = threadIdx.x;
  if (tid < IN_DIM) ls[tid] = 0.0f;
  __syncthreads();
  int i = blockIdx.x * blockDim.x + tid;
  if (i < n) {
    const float* row = agg + (size_t)i * IN_DIM;
    #pragma unroll
    for (int k = 0; k < IN_DIM; ++k) atomicAdd(&ls[k], row[k]);   // ds_add_f32
  }
  __syncthreads();
  if (tid < IN_DIM) atomicAdd(&s20[tid], ls[tid]);
}

// ---------------- Gram matrix G = agg^T agg via WMMA f32 16x16x4 -----------
// One wave accumulates the zero-padded 32x32 Gram. A-tile == B-tile by
// symmetry (4 nodes x 16 feats), so a single v2f load per feat-block feeds
// both operands. ISA layout (7.12.2): A 16x4 -> VGPR0 = K0|K2, VGPR1 = K1|K3
// (lane half selects K-pair); C/D 16x16 -> lane half selects M vs M+8.
// agg carries AGG_PAD zero rows, so node loads need no bounds checks; the
// feature-pad (feats 20..31) uses a clamped-address load + select.
__global__ void k_gram(const float* __restrict__ agg, float* __restrict__ G,
                       int n) {
  int lane = threadIdx.x & 31;
  int half = lane >> 4;            // 0: K0/K1 pair, 1: K2/K3 pair
  int lm   = lane & 15;            // feat (A rows / B cols) within tile
  int wave  = blockIdx.x * (blockDim.x >> 5) + (threadIdx.x >> 5);
  int nwave = gridDim.x * (blockDim.x >> 5);
  int steps = (n + 3) >> 2;        // 4 nodes per WMMA step (pad rows are 0)

  v8f a00 = {}; v8f a01 = {}; v8f a10 = {}; v8f a11 = {};
  const int  f0   = lm;            // 0..15  (always < IN_DIM)
  const bool f1ok = (lm + 16 < IN_DIM);
  const int  f1c  = f1ok ? (lm + 16) : 0;   // clamped address, no branch

  #pragma unroll 2
  for (int s = wave; s < steps; s += nwave) {
    const float* r0 = agg + (size_t)(4 * s + 2 * half) * IN_DIM;
    v2f p, q;
    p.x = r0[f0];
    p.y = r0[IN_DIM + f0];
    float q0 = r0[f1c];
    float q1 = r0[IN_DIM + f1c];
    q.x = f1ok ? q0 : 0.0f;
    q.y = f1ok ? q1 : 0.0f;
    // D = A * B + C ; feats block i x block j
    a00 = __builtin_amdgcn_wmma_f32_16x16x4_f32(false, p, false, p, (short)0, a00, false, false);
    a01 = __builtin_amdgcn_wmma_f32_16x16x4_f32(false, p, false, q, (short)0, a01, false, false);
    a10 = __builtin_amdgcn_wmma_f32_16x16x4_f32(false, q, false, p, (short)0, a10, false, false);
    a11 = __builtin_amdgcn_wmma_f32_16x16x4_f32(false, q, false, q, (short)0, a11, false, false);
  }

  #pragma unroll
  for (int r = 0; r < 8; ++r) {
    int rr = r + 8 * half;         // C/D layout: lane half selects M vs M+8
    atomicAdd(&G[(rr     ) * 32 + lm     ], a00[r]);
    atomicAdd(&G[(rr     ) * 32 + lm + 16], a01[r]);
    atomicAdd(&G[(rr + 16) * 32 + lm     ], a10[r]);
    atomicAdd(&G[(rr + 16) * 32 + lm + 16], a11[r]);
  }
}

// ---------------- per-channel BN stats: scale/shift ------------------------
// mean[c] = (s20 . w_c)/n ; E[h^2][c] = w_c^T G w_c / n ; var = E[h^2]-mean^2
__global__ void k_stats(const float* __restrict__ G,
                        const float* __restrict__ s20,
                        const float* __restrict__ W,      // (20,4096)
                        const float* __restrict__ gamma,
                        const float* __restrict__ beta,
                        float* __restrict__ scale, float* __restrict__ shift,
                        int n) {
  __shared__ float sG[IN_DIM * IN_DIM];
  __shared__ float ss[IN_DIM];
  int tid = threadIdx.x;
  for (int t = tid; t < IN_DIM * IN_DIM; t += blockDim.x)
    sG[t] = G[(t / IN_DIM) * 32 + (t % IN_DIM)];
  if (tid < IN_DIM) ss[tid] = s20[tid];
  __syncthreads();

  int c = blockIdx.x * blockDim.x + tid;
  if (c >= HIDDEN) return;
  float wc[IN_DIM];
  #pragma unroll
  for (int i = 0; i < IN_DIM; ++i) wc[i] = W[i * HIDDEN + c];

  float sum = 0.0f, qf = 0.0f;
  #pragma unroll
  for (int i = 0; i < IN_DIM; ++i) {
    sum += ss[i] * wc[i];
    float gi = 0.0f;
    #pragma unroll
    for (int j = 0; j < IN_DIM; ++j) gi += sG[i * IN_DIM + j] * wc[j];
    qf += wc[i] * gi;
  }
  float invn = 1.0f / (float)n;
  float mean = sum * invn;
  float var  = qf * invn - mean * mean;
  float sc   = gamma[c] * rsqrtf(var + BN_EPS);
  scale[c] = sc;
  shift[c] = beta[c] - mean * sc;
}

// ---------------- fold BN+linear into 20x2 matrix + const ------------------
// Wf[i][o] = sum_c W[i][c]*scale[c]*lin_w[c][o]
// cons[o]  = sum_c shift[c]*lin_w[c][o] + lin_b[o]
__global__ void k_fuse(const float* __restrict__ W,
                       const float* __restrict__ lin_w,   // (4096,2)
                       const float* __restrict__ lin_b,
                       const float* __restrict__ scale,
                       const float* __restrict__ shift,
                       float* __restrict__ Wf, float* __restrict__ cons) {
  __shared__ float acc[IN_DIM * OUT_DIM + OUT_DIM];       // 42
  int tid = threadIdx.x;
  if (tid < IN_DIM * OUT_DIM + OUT_DIM) acc[tid] = 0.0f;
  __syncthreads();

  float loc[IN_DIM * OUT_DIM + OUT_DIM];
  #pragma unroll
  for (int t = 0; t < IN_DIM * OUT_DIM + OUT_DIM; ++t) loc[t] = 0.0f;

  for (int c = tid; c < HIDDEN; c += blockDim.x) {
    float sc = scale[c], sh = shift[c];
    float l0 = lin_w[c * 2 + 0], l1 = lin_w[c * 2 + 1];
    #pragma unroll
    for (int i = 0; i < IN_DIM; ++i) {
      float w = W[i * HIDDEN + c] * sc;
      loc[i * 2 + 0] += w * l0;
      loc[i * 2 + 1] += w * l1;
    }
    loc[IN_DIM * 2 + 0] += sh * l0;
    loc[IN_DIM * 2 + 1] += sh * l1;
  }
  #pragma unroll
  for (int t = 0; t < IN_DIM * OUT_DIM + OUT_DIM; ++t) atomicAdd(&acc[t], loc[t]);
  __syncthreads();
  if (tid < IN_DIM * OUT_DIM) Wf[tid] = acc[tid];
  else if (tid < IN_DIM * OUT_DIM + OUT_DIM)
    cons[tid - IN_DIM * OUT_DIM] = acc[tid] + lin_b[tid - IN_DIM * OUT_DIM];
}

// ---------------- rsu_embedding = BN(agg[0] @ W) ---------------------------
__global__ void k_rsu(const float* __restrict__ agg,
                      const float* __restrict__ W,
                      const float* __restrict__ scale,
                      const float* __restrict__ shift,
                      float* __restrict__ out_rsu) {
  int c = blockIdx.x * blockDim.x + threadIdx.x;
  if (c >= HIDDEN) return;
  float s = 0.0f;
  #pragma unroll
  for (int i = 0; i < IN_DIM; ++i) s += agg[i] * W[i * HIDDEN + c];
  out_rsu[c] = s * scale[c] + shift[c];
}

// ---------------- action_prob = softmax(relu(agg @ Wf + cons)) -------------
__global__ void k_prob(const float* __restrict__ agg,
                       const float* __restrict__ Wf,
                       const float* __restrict__ cons,
                       float* __restrict__ out, int n) {
  int m = blockIdx.x * blockDim.x + threadIdx.x;
  if (m >= n) return;
  const float* a = agg + (size_t)m * IN_DIM;
  float l0 = cons[0], l1 = cons[1];
  #pragma unroll
  for (int k = 0; k < IN_DIM; ++k) {
    float v = a[k];
    l0 += v * Wf[k * 2 + 0];
    l1 += v * Wf[k * 2 + 1];
  }
  l0 = fmaxf(l0, 0.0f);
  l1 = fmaxf(l1, 0.0f);
  float mx = fmaxf(l0, l1);
  float e0 = expf(l0 - mx), e1 = expf(l1 - mx);
  float inv = 1.0f / (e0 + e1);
  out[2 * m + 0] = e0 * inv;
  out[2 * m + 1] = e1 * inv;
}

// ---------------------------------------------------------------------------
extern "C" void kernel_launch(void* const* d_in, const int* in_sizes, int n_in,
                              void* d_out, int out_size, void* d_ws, size_t ws_size,
                              hipStream_t stream) {
  const float* nf    = (const float*)d_in[0];
  const int*   ei    = (const int*)  d_in[1];
  const float* gcnW  = (const float*)d_in[2];
  // d_in[3] = gcn_b: cancels identically in batchnorm -> unused.
  const float* gamma = (const float*)d_in[4];
  const float* beta  = (const float*)d_in[5];
  const float* linW  = (const float*)d_in[6];
  const float* linB  = (const float*)d_in[7];
  float* out = (float*)d_out;

  const int n      = in_sizes[0] / IN_DIM;
  const int nedges = in_sizes[1] / 2;

  // workspace layout (floats)
  float* w     = (float*)d_ws;
  float* deg   = w;
  float* dinv  = deg  + n;
  float* agg   = dinv + n;                           // (n + AGG_PAD) * 20
  float* s20   = agg  + (size_t)(n + AGG_PAD) * IN_DIM;  // 32
  float* G     = s20  + 32;                          // 32*32
  float* scale = G    + 32 * 32;                     // 4096
  float* shift = scale + HIDDEN;                     // 4096
  float* Wf    = shift + HIDDEN;                     // 40
  float* cons  = Wf   + IN_DIM * OUT_DIM;            // 2

  const int B = 256;
  const int gN  = (n + B - 1) / B;
  const int gE  = (nedges + B - 1) / B;
  const int gNK = (n * IN_DIM + B - 1) / B;

  k_init   <<<gN, B, 0, stream>>>(deg, G, s20, agg + (size_t)n * IN_DIM, n);
  k_deg    <<<gE, B, 0, stream>>>(ei, deg, nedges);
  k_dinv   <<<gN, B, 0, stream>>>(deg, dinv, n);
  k_selfagg<<<gNK, B, 0, stream>>>(nf, dinv, agg, n * IN_DIM);
  k_edge   <<<gE, B, 0, stream>>>(ei, nf, dinv, agg, nedges);
  k_s20    <<<gN, B, 0, stream>>>(agg, s20, n);
  k_gram   <<<32, B, 0, stream>>>(agg, G, n);                 // 256 waves
  k_stats  <<<HIDDEN / B, B, 0, stream>>>(G, s20, gcnW, gamma, beta,
                                          scale, shift, n);
  k_fuse   <<<1, B, 0, stream>>>(gcnW, linW, linB, scale, shift, Wf, cons);
  k_rsu    <<<HIDDEN / B, B, 0, stream>>>(agg, gcnW, scale, shift,
                                          out + (size_t)2 * n);
  k_prob   <<<gN, B, 0, stream>>>(agg, Wf, cons, out, n);
}